// S6_20005957665390
// MI455X (gfx1250) — compile-verified
//
#include <hip/hip_runtime.h>
#include <hip/hip_bf16.h>
#include <math.h>

// ---------------- problem constants (from reference) ----------------
#define BATCH   4
#define SEQL    2048
#define DMODEL  512
#define NSTATE  16
#define OUTD    512
#define MROWS   (BATCH * SEQL)     // 8192 token rows
#define KDIM    512                // all GEMMs reduce over 512
#define NCHUNK  16                 // scan chunks along L
#define CLEN    (SEQL / NCHUNK)    // 128 timesteps per chunk
#define LN_EPS  1e-5f

typedef float v2f __attribute__((ext_vector_type(2)));
typedef float v8f __attribute__((ext_vector_type(8)));

// =====================================================================
// GEMM: out[M,Nout] = act(A[M,512] @ W[512,Nout] + bias)
// fp32 WMMA 16x16x4. Block = 128 threads = 4 waves; each wave owns one
// 16x16 output tile (4 consecutive M-tiles per block, one N-tile).
// The K x 16 weight panel is staged once in LDS (32 KB) and reused by
// all 4 waves for all 128 K-steps.
//
// Fragment layouts (CDNA5 ISA 7.12.2, f32):
//   A 16x4 : lane = m + 16*(k>=2), vgpr = k%2     -> per-lane float2 {k,k+1}
//   B 4x16 : lane = n + 16*(k>=2), vgpr = k%2     (symmetric striping)
//   C 16x16: lanes 0-15 vgpr r -> M=r ; lanes 16-31 vgpr r -> M=r+8
// =====================================================================
__global__ __launch_bounds__(128)
void s6_gemm_wmma_k512(const float* __restrict__ A,
                       const float* __restrict__ W,
                       const float* __restrict__ bias,
                       float* __restrict__ out,
                       int Nout, int act /*0=none,1=softplus*/) {
    __shared__ float sB[KDIM * 16];

    const int n0 = blockIdx.y << 4;
    // cooperative load of the K x 16 weight panel
    for (int idx = threadIdx.x; idx < KDIM * 16; idx += 128) {
        const int k = idx >> 4, nn = idx & 15;
        sB[idx] = W[(size_t)k * Nout + n0 + nn];
    }
    __syncthreads();

    const int wave  = threadIdx.x >> 5;
    const int lane  = threadIdx.x & 31;
    const int mtile = (blockIdx.x << 2) + wave;
    const int mrow  = (mtile << 4) + (lane & 15);   // A-frag row (m = lane%16)
    const int klo   = (lane >> 4) << 1;             // 0 or 2 (K sub-pair)
    const int ncol  = lane & 15;                    // B-frag column

    v8f acc = {0.f, 0.f, 0.f, 0.f, 0.f, 0.f, 0.f, 0.f};

    const float* ap = A + (size_t)mrow * KDIM + klo;
#pragma unroll 4
    for (int k0 = 0; k0 < KDIM; k0 += 4) {
        v2f av; av.x = ap[k0];  av.y = ap[k0 + 1];
        const int kk = (k0 + klo) << 4;
        v2f bv; bv.x = sB[kk + ncol]; bv.y = sB[kk + 16 + ncol];
        acc = __builtin_amdgcn_wmma_f32_16x16x4_f32(
            /*neg_a=*/false, av, /*neg_b=*/false, bv,
            /*c_mod=*/(short)0, acc, /*reuse_a=*/false, /*reuse_b=*/false);
    }

    const float bb  = bias[n0 + ncol];
    const int   mhi = (lane >> 4) << 3;   // lanes 16-31 hold rows 8..15
#pragma unroll
    for (int r = 0; r < 8; ++r) {
        float v = acc[r] + bb;
        if (act == 1) v = (v > 20.f) ? v : log1pf(expf(v));  // softplus
        out[(size_t)((mtile << 4) + mhi + r) * Nout + n0 + ncol] = v;
    }
}

// =====================================================================
// Scan pass 1: per chunk, compute total a-product and local end-state.
// grid = (D/256, NCHUNK, BATCH), block = 256 (one thread per d).
// =====================================================================
__global__ __launch_bounds__(256)
void s6_scan_pass1(const float* __restrict__ x,
                   const float* __restrict__ dt,
                   const float* __restrict__ Bm,
                   const float* __restrict__ A_log,
                   float* __restrict__ hend,
                   float* __restrict__ aend) {
    const int b = blockIdx.z, c = blockIdx.y;
    const int d = blockIdx.x * 256 + threadIdx.x;

    __shared__ float sBm[CLEN * NSTATE];
    for (int idx = threadIdx.x; idx < CLEN * NSTATE; idx += 256)
        sBm[idx] = Bm[((size_t)b * SEQL + c * CLEN) * NSTATE + idx];
    __syncthreads();

    float Ac[NSTATE];
#pragma unroll
    for (int n = 0; n < NSTATE; ++n) Ac[n] = -expf(A_log[d * NSTATE + n]);

    float h[NSTATE], ap[NSTATE];
#pragma unroll
    for (int n = 0; n < NSTATE; ++n) { h[n] = 0.f; ap[n] = 1.f; }

    for (int t = 0; t < CLEN; ++t) {
        const size_t ofs = ((size_t)b * SEQL + c * CLEN + t) * DMODEL + d;
        const float dtv = dt[ofs];
        const float ub  = dtv * x[ofs];
        const int   bmo = t * NSTATE;
#pragma unroll
        for (int n = 0; n < NSTATE; ++n) {
            const float a = __expf(dtv * Ac[n]);
            h[n]  = a * h[n] + ub * sBm[bmo + n];
            ap[n] *= a;
        }
    }

    const size_t base = (((size_t)b * DMODEL + d) * NCHUNK + c) * NSTATE;
#pragma unroll
    for (int n = 0; n < NSTATE; ++n) { hend[base + n] = h[n]; aend[base + n] = ap[n]; }
}

// =====================================================================
// Scan pass 2: sequential stitch of chunk carries per (b,d,n).
// 32768 threads, each walks 16 chunks.
// =====================================================================
__global__ __launch_bounds__(256)
void s6_scan_pass2(const float* __restrict__ hend,
                   const float* __restrict__ aend,
                   float* __restrict__ carry) {
    const int idx = blockIdx.x * 256 + threadIdx.x;   // < B*D*N = 32768
    const size_t bd = (size_t)(idx >> 4);
    const int n = idx & 15;
    float cv = 0.f;
    for (int c = 0; c < NCHUNK; ++c) {
        const size_t o = (bd * NCHUNK + c) * NSTATE + n;
        carry[o] = cv;
        cv = aend[o] * cv + hend[o];
    }
}

// =====================================================================
// Scan pass 3: re-run chunk with correct initial state, contract with
// Cm, add D-skip, exact-erf GELU. Writes y_gelu[B,L,D].
// =====================================================================
__global__ __launch_bounds__(256)
void s6_scan_pass3(const float* __restrict__ x,
                   const float* __restrict__ dt,
                   const float* __restrict__ Bm,
                   const float* __restrict__ Cm,
                   const float* __restrict__ A_log,
                   const float* __restrict__ carry,
                   const float* __restrict__ Dskip,
                   float* __restrict__ yout) {
    const int b = blockIdx.z, c = blockIdx.y;
    const int d = blockIdx.x * 256 + threadIdx.x;

    __shared__ float sBm[CLEN * NSTATE];
    __shared__ float sCm[CLEN * NSTATE];
    for (int idx = threadIdx.x; idx < CLEN * NSTATE; idx += 256) {
        const size_t g = ((size_t)b * SEQL + c * CLEN) * NSTATE + idx;
        sBm[idx] = Bm[g];
        sCm[idx] = Cm[g];
    }
    __syncthreads();

    float Ac[NSTATE];
#pragma unroll
    for (int n = 0; n < NSTATE; ++n) Ac[n] = -expf(A_log[d * NSTATE + n]);

    float h[NSTATE];
    const size_t base = (((size_t)b * DMODEL + d) * NCHUNK + c) * NSTATE;
#pragma unroll
    for (int n = 0; n < NSTATE; ++n) h[n] = carry[base + n];

    const float dsk = Dskip[d];
    for (int t = 0; t < CLEN; ++t) {
        const size_t ofs = ((size_t)b * SEQL + c * CLEN + t) * DMODEL + d;
        const float dtv = dt[ofs];
        const float xv  = x[ofs];
        const float ub  = dtv * xv;
        const int   so  = t * NSTATE;
        float y = 0.f;
#pragma unroll
        for (int n = 0; n < NSTATE; ++n) {
            const float a = __expf(dtv * Ac[n]);
            h[n] = a * h[n] + ub * sBm[so + n];
            y += sCm[so + n] * h[n];
        }
        y += dsk * xv;
        yout[ofs] = 0.5f * y * (1.f + erff(y * 0.70710678118654752f));  // exact GELU
    }
}

// =====================================================================
// LayerNorm + residual: one row (512 elems) per block of 256 threads.
// hout = hin + LN(ymix)*g + b     (bias was added in the GEMM epilogue)
// =====================================================================
__global__ __launch_bounds__(256)
void s6_ln_residual(const float* __restrict__ ymix,
                    const float* __restrict__ hin,
                    const float* __restrict__ g,
                    const float* __restrict__ be,
                    float* __restrict__ hout) {
    const size_t row = blockIdx.x;
    const int tid = threadIdx.x;
    const float v0 = ymix[row * DMODEL + tid];
    const float v1 = ymix[row * DMODEL + 256 + tid];

    __shared__ float s1[256], s2[256];
    s1[tid] = v0 + v1;
    s2[tid] = v0 * v0 + v1 * v1;
    __syncthreads();
    for (int st = 128; st > 0; st >>= 1) {
        if (tid < st) { s1[tid] += s1[tid + st]; s2[tid] += s2[tid + st]; }
        __syncthreads();
    }
    const float mu   = s1[0] * (1.f / DMODEL);
    const float var  = s2[0] * (1.f / DMODEL) - mu * mu;
    const float rstd = rsqrtf(var + LN_EPS);

    hout[row * DMODEL + tid] =
        hin[row * DMODEL + tid] + (v0 - mu) * rstd * g[tid] + be[tid];
    hout[row * DMODEL + 256 + tid] =
        hin[row * DMODEL + 256 + tid] + (v1 - mu) * rstd * g[256 + tid] + be[256 + tid];
}

// =====================================================================
// Host-side orchestration
// =====================================================================
extern "C" void kernel_launch(void* const* d_in, const int* in_sizes, int n_in,
                              void* d_out, int out_size, void* d_ws, size_t ws_size,
                              hipStream_t stream) {
    (void)in_sizes; (void)n_in; (void)out_size; (void)ws_size;

    const float* x      = (const float*)d_in[0];   // [B,L,D]
    const float* A_log  = (const float*)d_in[1];   // [2,D,N]
    const float* W_B    = (const float*)d_in[2];   // [2,D,N]
    const float* b_B    = (const float*)d_in[3];   // [2,N]
    const float* W_C    = (const float*)d_in[4];   // [2,D,N]
    const float* b_C    = (const float*)d_in[5];   // [2,N]
    const float* W_dt   = (const float*)d_in[6];   // [2,D,D]
    const float* b_dt   = (const float*)d_in[7];   // [2,D]
    const float* D_skip = (const float*)d_in[8];   // [2,D]
    const float* W_mix  = (const float*)d_in[9];   // [2,D,D]
    const float* b_mix  = (const float*)d_in[10];  // [2,D]
    const float* ln_g   = (const float*)d_in[11];  // [2,D]
    const float* ln_b   = (const float*)d_in[12];  // [2,D]
    const float* W_dec  = (const float*)d_in[13];  // [D,O]
    const float* b_dec  = (const float*)d_in[14];  // [O]
    float* out = (float*)d_out;                    // [B,L,O]

    // ---- workspace carve-up (floats) ----
    float* ws = (float*)d_ws;
    size_t off = 0;
    float* dtb   = ws + off; off += (size_t)MROWS * DMODEL;          // 16 MB
    float* Bmb   = ws + off; off += (size_t)MROWS * NSTATE;
    float* Cmb   = ws + off; off += (size_t)MROWS * NSTATE;
    float* ygelu = ws + off; off += (size_t)MROWS * DMODEL;          // 16 MB
    float* ymix  = ws + off; off += (size_t)MROWS * DMODEL;          // 16 MB
    float* h1    = ws + off; off += (size_t)MROWS * DMODEL;          // 16 MB
    float* h2    = ws + off; off += (size_t)MROWS * DMODEL;          // 16 MB
    float* hend  = ws + off; off += (size_t)BATCH * DMODEL * NCHUNK * NSTATE;
    float* aend  = ws + off; off += (size_t)BATCH * DMODEL * NCHUNK * NSTATE;
    float* carry = ws + off; off += (size_t)BATCH * DMODEL * NCHUNK * NSTATE;

    const int DN = DMODEL * NSTATE;
    const int DD = DMODEL * DMODEL;

    const dim3 gBig(MROWS / 64, DMODEL / 16);   // D->D GEMMs: 128 x 32 blocks
    const dim3 gBC (MROWS / 64, 1);             // D->N GEMMs
    const dim3 gScan(DMODEL / 256, NCHUNK, BATCH);

    for (int i = 0; i < 2; ++i) {
        const float* hin = (i == 0) ? x : h1;
        float* hout      = (i == 0) ? h1 : h2;

        // dt = softplus(hin @ W_dt + b_dt)
        s6_gemm_wmma_k512<<<gBig, 128, 0, stream>>>(
            hin, W_dt + (size_t)i * DD, b_dt + i * DMODEL, dtb, DMODEL, 1);
        // Bm, Cm projections (D -> 16)
        s6_gemm_wmma_k512<<<gBC, 128, 0, stream>>>(
            hin, W_B + (size_t)i * DN, b_B + i * NSTATE, Bmb, NSTATE, 0);
        s6_gemm_wmma_k512<<<gBC, 128, 0, stream>>>(
            hin, W_C + (size_t)i * DN, b_C + i * NSTATE, Cmb, NSTATE, 0);

        // chunked first-order scan
        s6_scan_pass1<<<gScan, 256, 0, stream>>>(
            hin, dtb, Bmb, A_log + (size_t)i * DN, hend, aend);
        s6_scan_pass2<<<BATCH * DMODEL * NSTATE / 256, 256, 0, stream>>>(
            hend, aend, carry);
        s6_scan_pass3<<<gScan, 256, 0, stream>>>(
            hin, dtb, Bmb, Cmb, A_log + (size_t)i * DN, carry,
            D_skip + i * DMODEL, ygelu);

        // mix GEMM (+bias), then LN + residual
        s6_gemm_wmma_k512<<<gBig, 128, 0, stream>>>(
            ygelu, W_mix + (size_t)i * DD, b_mix + i * DMODEL, ymix, DMODEL, 0);
        s6_ln_residual<<<MROWS, 256, 0, stream>>>(
            ymix, hin, ln_g + i * DMODEL, ln_b + i * DMODEL, hout);
    }

    // decoder: out = h2 @ W_dec + b_dec
    s6_gemm_wmma_k512<<<gBig, 128, 0, stream>>>(h2, W_dec, b_dec, out, OUTD, 0);
}